// MoE_36326833389779
// MI455X (gfx1250) — compile-verified
//
#include <hip/hip_runtime.h>
#include <math.h>

// Problem dims (match reference)
#define B_ 4096
#define D_ 1024
#define H_ 2048
#define O_ 1024
#define E_ 16
#define K_ 4

// LDS padding: row stride in floats chosen so byte stride % 256 == 8
// -> each of the 16 A-matrix rows lands on a distinct LDS bank pair for b64 loads.
#define XS_STRIDE 1026   // X tile rows (D_=1024 data + 2 pad)
#define HS_STRIDE 2050   // H tile rows (H_=2048 data + 2 pad)
#define SMEM_FLOATS (16 * XS_STRIDE + 16 * HS_STRIDE)
#define SMEM_BYTES (SMEM_FLOATS * 4)

typedef float v2f __attribute__((ext_vector_type(2)));
typedef float v8f __attribute__((ext_vector_type(8)));

// ---------------------------------------------------------------- zero init
__global__ __launch_bounds__(256) void moe_zero_kernel(float* __restrict__ y,
                                                       int n,
                                                       int* __restrict__ ecnt) {
  int i = blockIdx.x * 256 + threadIdx.x;
  if (i < n) y[i] = 0.f;
  if (i < E_) ecnt[i] = 0;
}

// ------------------------------------------------- gating: logits + stddev
// thread -> (token, expert); x broadcast across 16-thread groups, w_* coalesced.
__global__ __launch_bounds__(256) void moe_gate_kernel(
    const float* __restrict__ x, const float* __restrict__ noise,
    const float* __restrict__ w_gate, const float* __restrict__ w_noise,
    float* __restrict__ clean, float* __restrict__ sdv,
    float* __restrict__ noisy) {
  int idx = blockIdx.x * 256 + threadIdx.x;  // [0, B_*E_)
  int t = idx >> 4;
  int e = idx & 15;
  const float* xr = x + (size_t)t * D_;
  float ag = 0.f, an = 0.f;
  for (int d = 0; d < D_; ++d) {
    float xv = xr[d];
    ag = fmaf(xv, w_gate[d * E_ + e], ag);
    an = fmaf(xv, w_noise[d * E_ + e], an);
  }
  float sp = (an > 20.f) ? an : log1pf(expf(an));  // softplus
  float sd = sp + 0.01f;                           // + NOISE_EPS
  clean[idx] = ag;
  sdv[idx] = sd;
  noisy[idx] = ag + noise[idx] * sd;
}

// -------------------------------- routing: top-5, gates, normcdf, lists
__global__ __launch_bounds__(256) void moe_route_kernel(
    const float* __restrict__ clean, const float* __restrict__ sdv,
    const float* __restrict__ noisy, float* __restrict__ gates,
    float* __restrict__ prob, int* __restrict__ ecnt, int* __restrict__ etok,
    float* __restrict__ egate) {
  int t = blockIdx.x * 256 + threadIdx.x;
  if (t >= B_) return;
  float nl[E_], cl[E_], sd[E_];
#pragma unroll
  for (int e = 0; e < E_; ++e) {
    nl[e] = noisy[t * E_ + e];
    cl[e] = clean[t * E_ + e];
    sd[e] = sdv[t * E_ + e];
  }
  // top-(K+1) by repeated scan (first-occurrence ties, like lax.top_k)
  int mask = 0, sel[K_ + 1];
  float sv[K_ + 1];
  for (int j = 0; j < K_ + 1; ++j) {
    float best = -INFINITY;
    int bi = 0;
#pragma unroll
    for (int e = 0; e < E_; ++e)
      if (!((mask >> e) & 1) && nl[e] > best) { best = nl[e]; bi = e; }
    mask |= 1 << bi;
    sv[j] = best;
    sel[j] = bi;
  }
  // softmax over top-K (sv[0] is the max)
  float m = sv[0], sum = 0.f, g[K_];
#pragma unroll
  for (int k = 0; k < K_; ++k) { g[k] = expf(sv[k] - m); sum += g[k]; }
  float inv = 1.f / sum;
  float grow[E_];
#pragma unroll
  for (int e = 0; e < E_; ++e) grow[e] = 0.f;
#pragma unroll
  for (int k = 0; k < K_; ++k) grow[sel[k]] = g[k] * inv;

  float thr_in = sv[K_], thr_out = sv[K_ - 1];
  const float rs2 = 0.70710678118654752440f;
#pragma unroll
  for (int e = 0; e < E_; ++e) {
    gates[t * E_ + e] = grow[e];
    float thr = (nl[e] > thr_in) ? thr_in : thr_out;
    float z = (cl[e] - thr) / sd[e];
    prob[t * E_ + e] = 0.5f * (1.f + erff(z * rs2));  // norm_cdf
  }
#pragma unroll
  for (int k = 0; k < K_; ++k) {
    int e = sel[k];
    int pos = atomicAdd(&ecnt[e], 1);
    etok[e * B_ + pos] = t;
    egate[e * B_ + pos] = g[k] * inv;
  }
}

// ---------------------- deterministic column reductions: importance, load
__global__ __launch_bounds__(256) void moe_reduce_kernel(
    const float* __restrict__ gates, const float* __restrict__ prob,
    float* __restrict__ importance, float* __restrict__ load) {
  __shared__ float sg[256], sp[256];
  int e = blockIdx.x, tid = threadIdx.x;
  float a = 0.f, b = 0.f;
  for (int t = tid; t < B_; t += 256) {
    a += gates[t * E_ + e];
    b += prob[t * E_ + e];
  }
  sg[tid] = a;
  sp[tid] = b;
  __syncthreads();
  for (int s = 128; s; s >>= 1) {
    if (tid < s) { sg[tid] += sg[tid + s]; sp[tid] += sp[tid + s]; }
    __syncthreads();
  }
  if (tid == 0) { importance[e] = sg[0]; load[e] = sp[0]; }
}

// ----------------------------------------------------------- cv^2 loss
__global__ void moe_loss_kernel(const float* __restrict__ importance,
                                const float* __restrict__ load,
                                float* __restrict__ loss_out) {
  if (threadIdx.x != 0) return;
  float mi = 0.f, ml = 0.f;
  for (int e = 0; e < E_; ++e) { mi += importance[e]; ml += load[e]; }
  mi *= (1.f / E_);
  ml *= (1.f / E_);
  float vi = 0.f, vl = 0.f;
  for (int e = 0; e < E_; ++e) {
    float a = importance[e] - mi, b = load[e] - ml;
    vi += a * a;
    vl += b * b;
  }
  vi *= (1.f / (E_ - 1));  // ddof=1
  vl *= (1.f / (E_ - 1));
  float cvi = vi / (mi * mi + 1e-10f);
  float cvl = vl / (ml * ml + 1e-10f);
  *loss_out = (cvi + cvl) * 0.01f;  // LOSS_COEF
}

// -------------------------------------------------- fused sparse expert FFN
// grid = (256 tiles, 16 experts); block = 256 (8 wave32). One block handles a
// 16-token tile of one expert entirely in LDS:
//   Xs(16x1024) --WMMA_f32--> Hs(16x2048,relu) --WMMA_f32--> Outs(16x1024)
//   --softmax--> y += gate*exp(prob)  (global fp32 atomics)
__global__ __launch_bounds__(256) void moe_expert_kernel(
    const float* __restrict__ x, const float* __restrict__ W1,
    const float* __restrict__ b1, const float* __restrict__ W2,
    const float* __restrict__ b2, const int* __restrict__ etok,
    const float* __restrict__ egate, const int* __restrict__ ecnt,
    float* __restrict__ y) {
  const int e = blockIdx.y;
  const int cnt = ecnt[e];
  const int row0 = blockIdx.x * 16;
  if (row0 >= cnt) return;  // uniform exit: EXEC stays all-ones for WMMA

  extern __shared__ float smem[];
  float* Xs = smem;                       // 16 x XS_STRIDE
  float* Hs = smem + 16 * XS_STRIDE;      // 16 x HS_STRIDE
  float* Outs = smem;                     // 16 x O_ (reuses Xs region)
  __shared__ int s_tok[16];
  __shared__ float s_gate[16];

  const int tid = threadIdx.x;
  if (tid < 16) {
    int r = row0 + tid;
    s_tok[tid] = (r < cnt) ? etok[e * B_ + r] : -1;
    s_gate[tid] = (r < cnt) ? egate[e * B_ + r] : 0.f;
  }
  __syncthreads();

  // stage gathered X rows (coalesced; zero-fill padded rows)
  for (int i = tid; i < 16 * D_; i += 256) {
    int r = i >> 10, d = i & (D_ - 1);
    int t = s_tok[r];
    Xs[r * XS_STRIDE + d] = (t >= 0) ? x[(size_t)t * D_ + d] : 0.f;
  }
  __syncthreads();

  const int lane = tid & 31, wave = tid >> 5;
  const int nsel = lane & 15;            // N (and A-row M) index
  const int kHalf = (lane >> 4) << 1;    // 0 or 2 : K sub-pair
  const int mBase = (lane >> 4) << 3;    // 0 or 8 : C/D row base
  const float* W1e = W1 + (size_t)e * D_ * H_;
  const float* W2e = W2 + (size_t)e * H_ * O_;

  // ---- GEMM1: Hs = relu(Xs @ W1e + b1[e]) ; 128 N-tiles over 8 waves
  for (int nt = wave; nt < H_ / 16; nt += 8) {
    const int col = nt * 16 + nsel;
    const float* bp = W1e + col;
    v8f acc = {0.f, 0.f, 0.f, 0.f, 0.f, 0.f, 0.f, 0.f};
    for (int k = 0; k < D_; k += 4) {
      v2f a = *(const v2f*)&Xs[nsel * XS_STRIDE + k + kHalf];  // ds_load_b64
      v2f b;
      b.x = bp[(size_t)(k + kHalf) * H_];
      b.y = bp[(size_t)(k + kHalf + 1) * H_];
      acc = __builtin_amdgcn_wmma_f32_16x16x4_f32(false, a, false, b,
                                                  (short)0, acc, false, false);
    }
    float bias = b1[e * H_ + col];
#pragma unroll
    for (int v = 0; v < 8; ++v)
      Hs[(mBase + v) * HS_STRIDE + col] = fmaxf(acc[v] + bias, 0.f);
  }
  __syncthreads();

  // ---- GEMM2: Outs = Hs @ W2e + b2[e] ; 64 N-tiles over 8 waves
  for (int nt = wave; nt < O_ / 16; nt += 8) {
    const int col = nt * 16 + nsel;
    const float* bp = W2e + col;
    v8f acc = {0.f, 0.f, 0.f, 0.f, 0.f, 0.f, 0.f, 0.f};
    for (int k = 0; k < H_; k += 4) {
      v2f a = *(const v2f*)&Hs[nsel * HS_STRIDE + k + kHalf];
      v2f b;
      b.x = bp[(size_t)(k + kHalf) * O_];
      b.y = bp[(size_t)(k + kHalf + 1) * O_];
      acc = __builtin_amdgcn_wmma_f32_16x16x4_f32(false, a, false, b,
                                                  (short)0, acc, false, false);
    }
    float bias = b2[e * O_ + col];
#pragma unroll
    for (int v = 0; v < 8; ++v) Outs[(mBase + v) * O_ + col] = acc[v] + bias;
  }
  __syncthreads();

  // ---- per-row softmax, then y += gate * exp(prob)
  for (int r = wave; r < 16; r += 8) {
    int t = s_tok[r];
    if (t < 0) continue;
    float g = s_gate[r];
    const float* orow = &Outs[r * O_];
    float m = -INFINITY;
    for (int o = lane; o < O_; o += 32) m = fmaxf(m, orow[o]);
    for (int off = 16; off; off >>= 1) m = fmaxf(m, __shfl_xor(m, off, 32));
    float s = 0.f;
    for (int o = lane; o < O_; o += 32) s += expf(orow[o] - m);
    for (int off = 16; off; off >>= 1) s += __shfl_xor(s, off, 32);
    float inv = 1.f / s;
    float* yrow = y + (size_t)t * O_;
    for (int o = lane; o < O_; o += 32) {
      float p = expf(orow[o] - m) * inv;  // softmax prob
      atomicAdd(&yrow[o], g * expf(p));   // y = sum_e gate * exp(prob)
    }
  }
}

// ------------------------------------------------------------- launcher
extern "C" void kernel_launch(void* const* d_in, const int* in_sizes, int n_in,
                              void* d_out, int out_size, void* d_ws,
                              size_t ws_size, hipStream_t stream) {
  (void)in_sizes; (void)n_in; (void)out_size; (void)ws_size;
  const float* x = (const float*)d_in[0];
  const float* noise = (const float*)d_in[1];
  const float* w_gate = (const float*)d_in[2];
  const float* w_noise = (const float*)d_in[3];
  const float* W1 = (const float*)d_in[4];
  const float* b1 = (const float*)d_in[5];
  const float* W2 = (const float*)d_in[6];
  const float* b2 = (const float*)d_in[7];

  float* y = (float*)d_out;                       // B_*O_ floats
  float* loss = y + (size_t)B_ * O_;              // +1 float

  // workspace layout (256B aligned slices)
  char* ws = (char*)d_ws;
  const size_t BE = (size_t)B_ * E_ * sizeof(float);  // 256 KB
  float* clean = (float*)(ws);
  float* sdv = (float*)(ws + 1 * BE);
  float* noisy = (float*)(ws + 2 * BE);
  float* gates = (float*)(ws + 3 * BE);
  float* prob = (float*)(ws + 4 * BE);
  float* egate = (float*)(ws + 5 * BE);
  int* etok = (int*)(ws + 6 * BE);
  int* ecnt = (int*)(ws + 7 * BE);
  float* importance = (float*)(ws + 7 * BE + 256);
  float* load = (float*)(ws + 7 * BE + 512);

  // allow 192KB+ dynamic LDS on the expert kernel (320KB/WGP on CDNA5)
  (void)hipFuncSetAttribute(reinterpret_cast<const void*>(moe_expert_kernel),
                            hipFuncAttributeMaxDynamicSharedMemorySize,
                            SMEM_BYTES);

  const int NY = B_ * O_;
  moe_zero_kernel<<<(NY + 255) / 256, 256, 0, stream>>>(y, NY, ecnt);
  moe_gate_kernel<<<(B_ * E_) / 256, 256, 0, stream>>>(x, noise, w_gate,
                                                       w_noise, clean, sdv,
                                                       noisy);
  moe_route_kernel<<<(B_ + 255) / 256, 256, 0, stream>>>(
      clean, sdv, noisy, gates, prob, ecnt, etok, egate);
  moe_reduce_kernel<<<E_, 256, 0, stream>>>(gates, prob, importance, load);
  moe_loss_kernel<<<1, 32, 0, stream>>>(importance, load, loss);
  moe_expert_kernel<<<dim3(B_ / 16, E_), 256, SMEM_BYTES, stream>>>(
      x, W1, b1, W2, b2, etok, egate, ecnt, y);
}